// DogShitNet69_41661182771795
// MI455X (gfx1250) — compile-verified
//
#include <hip/hip_runtime.h>

typedef __attribute__((ext_vector_type(2))) float v2f;
typedef __attribute__((ext_vector_type(8))) float v8f;

#define NB 8           // batch
#define EPS_F 1e-5f

// ---------------------------------------------------------------------------
// mask downsample: out[p] = max over 2x2x2 children (reduce_window 2,2,2 s2)
// ---------------------------------------------------------------------------
template <typename MT, int DOUT>
__global__ void down_mask(const MT* __restrict__ mi, float* __restrict__ mo) {
  constexpr int NV = NB * DOUT * DOUT * DOUT;
  constexpr int DIN = DOUT * 2;
  int i = blockIdx.x * blockDim.x + threadIdx.x;
  if (i >= NV) return;
  int x = i & (DOUT - 1);
  int y = (i / DOUT) & (DOUT - 1);
  int z = (i / (DOUT * DOUT)) & (DOUT - 1);
  int b = i / (DOUT * DOUT * DOUT);
  float m = 0.f;
#pragma unroll
  for (int dz = 0; dz < 2; ++dz)
#pragma unroll
    for (int dy = 0; dy < 2; ++dy)
#pragma unroll
      for (int dx = 0; dx < 2; ++dx) {
        int idx = ((b * DIN + (2 * z + dz)) * DIN + (2 * y + dy)) * DIN + (2 * x + dx);
        m = fmaxf(m, (float)mi[idx]);
      }
  mo[i] = m;
}

// ---------------------------------------------------------------------------
// direct conv for the bandwidth-bound small-channel layers (0,1,2)
// one thread = one output voxel, all COUT channels in registers
// ---------------------------------------------------------------------------
template <int CIN, int COUT, int STRIDE, int DIN, int DOUT>
__global__ void conv_direct(const float* __restrict__ in, const float* __restrict__ w,
                            const int* __restrict__ inMaskI,
                            const float* __restrict__ outMaskF, const int* __restrict__ outMaskI,
                            float* __restrict__ out) {
  constexpr int NV = NB * DOUT * DOUT * DOUT;
  int i = blockIdx.x * blockDim.x + threadIdx.x;
  if (i >= NV) return;
  int xo = i & (DOUT - 1);
  int yo = (i / DOUT) & (DOUT - 1);
  int zo = (i / (DOUT * DOUT)) & (DOUT - 1);
  int b  = i / (DOUT * DOUT * DOUT);

  float acc[COUT];
#pragma unroll
  for (int c = 0; c < COUT; ++c) acc[c] = 0.f;

  for (int kz = 0; kz < 3; ++kz) {
    int zi = zo * STRIDE + kz - 1;
    if ((unsigned)zi >= (unsigned)DIN) continue;
    for (int ky = 0; ky < 3; ++ky) {
      int yi = yo * STRIDE + ky - 1;
      if ((unsigned)yi >= (unsigned)DIN) continue;
      for (int kx = 0; kx < 3; ++kx) {
        int xi = xo * STRIDE + kx - 1;
        if ((unsigned)xi >= (unsigned)DIN) continue;
        int vIdx = ((b * DIN + zi) * DIN + yi) * DIN + xi;
        int tap = (kz * 3 + ky) * 3 + kx;
        float msc = inMaskI ? (float)inMaskI[vIdx] : 1.0f;
#pragma unroll
        for (int ci = 0; ci < CIN; ++ci) {
          float xv = in[(size_t)vIdx * CIN + ci] * msc;
          const float* wp = w + (size_t)(tap * CIN + ci) * COUT;
#pragma unroll
          for (int co = 0; co < COUT; ++co) acc[co] = fmaf(xv, wp[co], acc[co]);
        }
      }
    }
  }
  float mo = outMaskF ? outMaskF[i] : (float)outMaskI[i];
#pragma unroll
  for (int co = 0; co < COUT; ++co) out[(size_t)i * COUT + co] = acc[co] * mo;
}

// ---------------------------------------------------------------------------
// implicit-GEMM conv using V_WMMA_F32_16X16X4_F32 for channel-rich layers.
// wave tile: M = 16 output voxels, N = 16 output channels, K = 27*CIN.
// Loop over the 27 taps (bounds check + base address hoisted per tap),
// unrolled inner loop over cin in chunks of 4 -> CIN/4 WMMAs per tap body.
// A frag (16x4 f32): lanes 0-15 hold (M=lane, K=kk+{0,1}); lanes 16-31 hold
// (M=lane-16, K=kk+{2,3}). B frag (4x16) symmetric. NVOX and COUT are
// multiples of 16 for all instantiations -> no predication, EXEC stays full.
// ---------------------------------------------------------------------------
template <int CIN, int COUT, int STRIDE, int DIN, int DOUT>
__global__ void conv_wmma(const float* __restrict__ in, const float* __restrict__ w,
                          const float* __restrict__ mout, float* __restrict__ out) {
  constexpr int NT = COUT / 16;
  constexpr int NVOX = NB * DOUT * DOUT * DOUT;
  constexpr int TOTAL_WAVES = (NVOX / 16) * NT;
  int wave = (int)((blockIdx.x * (unsigned)blockDim.x + threadIdx.x) >> 5);
  int lane = threadIdx.x & 31;
  if (wave >= TOTAL_WAVES) return;  // uniform per-wave branch

  int tileN = wave % NT;
  int tileM = wave / NT;
  int lhalf = lane >> 4;  // 0: K pair {0,1}; 1: K pair {2,3}
  int l16 = lane & 15;

  // voxel handled by this lane's A rows (DOUT is a power of two -> shifts)
  int vA = tileM * 16 + l16;
  int xo = vA & (DOUT - 1);
  int yo = (vA / DOUT) & (DOUT - 1);
  int zo = (vA / (DOUT * DOUT)) & (DOUT - 1);
  int b  = vA / (DOUT * DOUT * DOUT);

  int nCol = tileN * 16 + l16;  // B column for this lane

  v8f acc = {0.f, 0.f, 0.f, 0.f, 0.f, 0.f, 0.f, 0.f};

  for (int kz = 0; kz < 3; ++kz) {
    int zi = zo * STRIDE + kz - 1;
    for (int ky = 0; ky < 3; ++ky) {
      int yi = yo * STRIDE + ky - 1;
      for (int kx = 0; kx < 3; ++kx) {
        int xi = xo * STRIDE + kx - 1;
        bool ok = (unsigned)zi < (unsigned)DIN && (unsigned)yi < (unsigned)DIN &&
                  (unsigned)xi < (unsigned)DIN;
        const float* ain =
            in + (size_t)((b * DIN + zi) * DIN + yi) * DIN * CIN + (size_t)xi * CIN;
        int tap = (kz * 3 + ky) * 3 + kx;
        const float* wrow = w + (size_t)tap * CIN * COUT + nCol;
#pragma unroll
        for (int cc = 0; cc < CIN; cc += 4) {
          int c0 = cc + 2 * lhalf;  // this lane's K pair within the chunk
          v2f av = {ok ? ain[c0] : 0.f, ok ? ain[c0 + 1] : 0.f};
          v2f bv = {wrow[(size_t)c0 * COUT], wrow[(size_t)(c0 + 1) * COUT]};
          // D = A(16x4) * B(4x16) + C ; exact fp32, matches reference precision
          acc = __builtin_amdgcn_wmma_f32_16x16x4_f32(false, av, false, bv,
                                                      (short)0, acc, false, false);
        }
      }
    }
  }

  // C/D layout: VGPR r -> M = r + 8*lhalf, N = l16
#pragma unroll
  for (int r = 0; r < 8; ++r) {
    int vv = tileM * 16 + r + 8 * lhalf;
    out[(size_t)vv * COUT + tileN * 16 + l16] = acc[r] * mout[vv];
  }
}

// ---------------------------------------------------------------------------
// per-channel sum / sumsq / active count -> stats[0]=cnt, [1..C]=sum, [1+C..2C]=sumsq
// ---------------------------------------------------------------------------
__global__ void zero_f(float* __restrict__ p, int n) {
  int i = blockIdx.x * blockDim.x + threadIdx.x;
  if (i < n) p[i] = 0.f;
}

template <int C, typename MT>
__global__ void reduce_stats(const float* __restrict__ x, const MT* __restrict__ m,
                             int nvox, float* __restrict__ stats) {
  int c = blockIdx.y;
  float s = 0.f, s2 = 0.f, cnt = 0.f;
  for (int v = blockIdx.x * blockDim.x + threadIdx.x; v < nvox;
       v += gridDim.x * blockDim.x) {
    float xv = x[(size_t)v * C + c];
    s += xv;
    s2 += xv * xv;
    if (c == 0) cnt += (float)m[v];
  }
  __shared__ float ls[256], ls2[256], lc[256];
  int tid = threadIdx.x;
  ls[tid] = s; ls2[tid] = s2; lc[tid] = cnt;
  __syncthreads();
  for (int off = 128; off > 0; off >>= 1) {
    if (tid < off) {
      ls[tid] += ls[tid + off];
      ls2[tid] += ls2[tid + off];
      lc[tid] += lc[tid + off];
    }
    __syncthreads();
  }
  if (tid == 0) {
    atomicAdd(&stats[1 + c], ls[0]);
    atomicAdd(&stats[1 + C + c], ls2[0]);
    if (c == 0) atomicAdd(&stats[0], lc[0]);
  }
}

template <int C, typename MT>
__global__ void bn_relu(float* __restrict__ x, const MT* __restrict__ m,
                        const float* __restrict__ g, const float* __restrict__ bb,
                        const float* __restrict__ stats, int nvox) {
  int i = blockIdx.x * blockDim.x + threadIdx.x;
  if (i >= nvox * C) return;
  int v = i / C, c = i % C;
  float cnt = fmaxf(stats[0], 1.0f);
  float mean = stats[1 + c] / cnt;
  float var = stats[1 + C + c] / cnt - mean * mean;
  float y = (x[i] - mean) * (g[c] * rsqrtf(var + EPS_F)) + bb[c];
  x[i] = fmaxf(y, 0.f) * (float)m[v];
}

// ---------------------------------------------------------------------------
// sparse max-pool k=3 s=2 pad=1 on the 4^3 grid -> [8,2,2,2,64]
// ---------------------------------------------------------------------------
__global__ void final_pool(const float* __restrict__ x, const float* __restrict__ m5,
                           float* __restrict__ out) {
  int i = blockIdx.x * blockDim.x + threadIdx.x;
  if (i >= NB * 8 * 64) return;
  int c = i % 64; int v = i / 64;
  int xo = v & 1;
  int yo = (v >> 1) & 1;
  int zo = (v >> 2) & 1;
  int b  = v >> 3;

  float mo = 0.f;
#pragma unroll
  for (int dz = 0; dz < 2; ++dz)
#pragma unroll
    for (int dy = 0; dy < 2; ++dy)
#pragma unroll
      for (int dx = 0; dx < 2; ++dx)
        mo = fmaxf(mo, m5[((b * 4 + (2 * zo + dz)) * 4 + (2 * yo + dy)) * 4 + (2 * xo + dx)]);

  float best = 0.f;  // post-ReLU values are >= 0
  for (int kz = 0; kz < 3; ++kz) {
    int zi = 2 * zo + kz - 1; if ((unsigned)zi >= 4u) continue;
    for (int ky = 0; ky < 3; ++ky) {
      int yi = 2 * yo + ky - 1; if ((unsigned)yi >= 4u) continue;
      for (int kx = 0; kx < 3; ++kx) {
        int xi = 2 * xo + kx - 1; if ((unsigned)xi >= 4u) continue;
        int idx = ((b * 4 + zi) * 4 + yi) * 4 + xi;
        if (m5[idx] > 0.f) best = fmaxf(best, x[(size_t)idx * 64 + c]);
      }
    }
  }
  out[i] = (mo > 0.f) ? best : 0.f;
}

// ---------------------------------------------------------------------------
static inline int cdiv(long long a, long long b) { return (int)((a + b - 1) / b); }
static inline int imin(int a, int b) { return a < b ? a : b; }

extern "C" void kernel_launch(void* const* d_in, const int* in_sizes, int n_in,
                              void* d_out, int out_size, void* d_ws, size_t ws_size,
                              hipStream_t stream) {
  (void)in_sizes; (void)n_in; (void)out_size; (void)ws_size;
  const float* x_feats = (const float*)d_in[0];
  const int*   x_mask  = (const int*)d_in[1];
  const float *w[6], *g[6], *bb[6];
  for (int i = 0; i < 6; ++i) {
    w[i]  = (const float*)d_in[2 + 3 * i];
    g[i]  = (const float*)d_in[3 + 3 * i];
    bb[i] = (const float*)d_in[4 + 3 * i];
  }

  const int nv0 = NB * 128 * 128 * 128;
  const int nv1 = NB * 64 * 64 * 64;
  const int nv2 = NB * 32 * 32 * 32;
  const int nv3 = NB * 16 * 16 * 16;
  const int nv4 = NB * 8 * 8 * 8;
  const int nv5 = NB * 4 * 4 * 4;

  char* ws = (char*)d_ws;
  size_t off = 0;
  auto alloc = [&](size_t elems) {
    float* p = (float*)(ws + off);
    off += (elems * sizeof(float) + 255) & ~(size_t)255;
    return p;
  };
  float* buf0 = alloc((size_t)nv0 * 2);
  float* buf1 = alloc((size_t)nv1 * 4);
  float* buf2 = alloc((size_t)nv2 * 8);
  float* buf3 = alloc((size_t)nv3 * 16);
  float* buf4 = alloc((size_t)nv4 * 32);
  float* buf5 = alloc((size_t)nv5 * 64);
  float* m1 = alloc(nv1);
  float* m2 = alloc(nv2);
  float* m3 = alloc(nv3);
  float* m4 = alloc(nv4);
  float* m5 = alloc(nv5);
  float* stats = alloc(129);

  const int TB = 256;

  // mask pyramid
  down_mask<int, 64><<<cdiv(nv1, TB), TB, 0, stream>>>(x_mask, m1);
  down_mask<float, 32><<<cdiv(nv2, TB), TB, 0, stream>>>(m1, m2);
  down_mask<float, 16><<<cdiv(nv3, TB), TB, 0, stream>>>(m2, m3);
  down_mask<float, 8><<<cdiv(nv4, TB), TB, 0, stream>>>(m3, m4);
  down_mask<float, 4><<<cdiv(nv5, TB), TB, 0, stream>>>(m4, m5);

  // ---- layer 0: direct conv (bandwidth bound), int mask in & out ----
  conv_direct<1, 2, 1, 128, 128><<<cdiv(nv0, TB), TB, 0, stream>>>(
      x_feats, w[0], x_mask, nullptr, x_mask, buf0);
  zero_f<<<1, 256, 0, stream>>>(stats, 129);
  { dim3 gr(imin(2048, cdiv(nv0, TB)), 2);
    reduce_stats<2, int><<<gr, TB, 0, stream>>>(buf0, x_mask, nv0, stats); }
  bn_relu<2, int><<<cdiv((long long)nv0 * 2, TB), TB, 0, stream>>>(
      buf0, x_mask, g[0], bb[0], stats, nv0);

  // ---- layer 1 ----
  conv_direct<2, 4, 2, 128, 64><<<cdiv(nv1, TB), TB, 0, stream>>>(
      buf0, w[1], nullptr, m1, nullptr, buf1);
  zero_f<<<1, 256, 0, stream>>>(stats, 129);
  { dim3 gr(imin(2048, cdiv(nv1, TB)), 4);
    reduce_stats<4, float><<<gr, TB, 0, stream>>>(buf1, m1, nv1, stats); }
  bn_relu<4, float><<<cdiv((long long)nv1 * 4, TB), TB, 0, stream>>>(
      buf1, m1, g[1], bb[1], stats, nv1);

  // ---- layer 2 ----
  conv_direct<4, 8, 2, 64, 32><<<cdiv(nv2, TB), TB, 0, stream>>>(
      buf1, w[2], nullptr, m2, nullptr, buf2);
  zero_f<<<1, 256, 0, stream>>>(stats, 129);
  { dim3 gr(imin(2048, cdiv(nv2, TB)), 8);
    reduce_stats<8, float><<<gr, TB, 0, stream>>>(buf2, m2, nv2, stats); }
  bn_relu<8, float><<<cdiv((long long)nv2 * 8, TB), TB, 0, stream>>>(
      buf2, m2, g[2], bb[2], stats, nv2);

  // ---- layer 3: WMMA implicit GEMM (K=216, N=16) ----
  { long long waves = (long long)(nv3 / 16) * (16 / 16);
    conv_wmma<8, 16, 2, 32, 16><<<cdiv(waves * 32, TB), TB, 0, stream>>>(
        buf2, w[3], m3, buf3); }
  zero_f<<<1, 256, 0, stream>>>(stats, 129);
  { dim3 gr(imin(2048, cdiv(nv3, TB)), 16);
    reduce_stats<16, float><<<gr, TB, 0, stream>>>(buf3, m3, nv3, stats); }
  bn_relu<16, float><<<cdiv((long long)nv3 * 16, TB), TB, 0, stream>>>(
      buf3, m3, g[3], bb[3], stats, nv3);

  // ---- layer 4: WMMA (K=432, N=32) ----
  { long long waves = (long long)(nv4 / 16) * (32 / 16);
    conv_wmma<16, 32, 2, 16, 8><<<cdiv(waves * 32, TB), TB, 0, stream>>>(
        buf3, w[4], m4, buf4); }
  zero_f<<<1, 256, 0, stream>>>(stats, 129);
  { dim3 gr(imin(2048, cdiv(nv4, TB)), 32);
    reduce_stats<32, float><<<gr, TB, 0, stream>>>(buf4, m4, nv4, stats); }
  bn_relu<32, float><<<cdiv((long long)nv4 * 32, TB), TB, 0, stream>>>(
      buf4, m4, g[4], bb[4], stats, nv4);

  // ---- layer 5: WMMA (K=864, N=64) ----
  { long long waves = (long long)(nv5 / 16) * (64 / 16);
    conv_wmma<32, 64, 2, 8, 4><<<cdiv(waves * 32, TB), TB, 0, stream>>>(
        buf4, w[5], m5, buf5); }
  zero_f<<<1, 256, 0, stream>>>(stats, 129);
  { dim3 gr(imin(2048, cdiv(nv5, TB)), 64);
    reduce_stats<64, float><<<gr, TB, 0, stream>>>(buf5, m5, nv5, stats); }
  bn_relu<64, float><<<cdiv((long long)nv5 * 64, TB), TB, 0, stream>>>(
      buf5, m5, g[5], bb[5], stats, nv5);

  // ---- final sparse max pool ----
  final_pool<<<cdiv(NB * 8 * 64, TB), TB, 0, stream>>>(buf5, m5, (float*)d_out);
}